// NodeEncoder_35699768164381
// MI455X (gfx1250) — compile-verified
//
#include <hip/hip_runtime.h>

#define NNODES 50000
#define NEDGES 800000
#define FDIM   128
#define NEG_SLOPE 0.2f

typedef float v2f __attribute__((ext_vector_type(2)));
typedef float v8f __attribute__((ext_vector_type(8)));

__device__ __forceinline__ float lrelu(float x) { return x > 0.0f ? x : NEG_SLOPE * x; }

// ---------------------------------------------------------------------------
// Generic WMMA fp32 GEMM:  C[M x NC] = A[M x K](LDA) @ W[K x NC] (+ bias)
// Block = 256 threads = 8 waves (wave32). Each wave owns a 16x64 output tile.
// W is staged through LDS in KTILE-row chunks (KTILE*NC*4 bytes == 64KB).
// Uses V_WMMA_F32_16X16X4_F32; per ISA layouts:
//   A 16x4 : v0 = A[m][k0], v1 = A[m][k0+1], k0 = 0 (lanes 0-15) / 2 (16-31)
//   B 4x16 : v0 = B[k0][n], v1 = B[k0+1][n]
//   C 16x16: VGPR v -> row v (lanes 0-15) / row v+8 (lanes 16-31), col = lane%16
// K, LDA, LDC are compile-time so row strides fold into instruction offsets.
// M must be a multiple of 16; NC a multiple of 64; K a multiple of KTILE.
// ---------------------------------------------------------------------------
template <int NC, int KTILE, int K, int LDA, int LDC>
__global__ __launch_bounds__(256)
void gemm_wmma(const float* __restrict__ A,
               const float* __restrict__ W,
               const float* __restrict__ bias,
               float* __restrict__ C,
               int M) {
  __shared__ float sW[KTILE * NC];

  const int lane = threadIdx.x & 31;
  const int wave = threadIdx.x >> 5;
  const int lm   = lane & 15;
  const int kb   = (lane >> 4) << 1;   // 0 or 2 (K sub-offset of this half-wave)
  const int rO   = (lane >> 4) << 3;   // 0 or 8 (row offset in C fragment)

  constexpr int tilesN = NC >> 6;
  const int tilesM = M >> 4;
  const int tile   = blockIdx.x * 8 + wave;
  const bool active = tile < tilesM * tilesN;
  const int tm = active ? (tile / tilesN) : 0;
  const int tn = active ? (tile % tilesN) : 0;
  const int row0 = tm << 4;
  const int col0 = tn << 6;

  v8f acc[4];
#pragma unroll
  for (int s = 0; s < 4; ++s) {
    float bv = bias ? bias[col0 + s * 16 + lm] : 0.0f;
#pragma unroll
    for (int j = 0; j < 8; ++j) acc[s][j] = bv;
  }

  // A fragment pointer for this lane; advances by 4 floats per K-step
  const float* aRow = A + (size_t)(row0 + lm) * LDA + kb;

  constexpr int nChunks = K / KTILE;
#pragma unroll 1
  for (int kc = 0; kc < nChunks; ++kc) {
    // cooperative stage of one K-chunk of W into LDS
    const float4* gsrc = reinterpret_cast<const float4*>(W + (size_t)kc * KTILE * NC);
    float4* ldst = reinterpret_cast<float4*>(sW);
#pragma unroll
    for (int i = 0; i < (KTILE * NC) / 4 / 256; ++i)
      ldst[threadIdx.x + i * 256] = gsrc[threadIdx.x + i * 256];
    __syncthreads();

    if (active) {  // wave-uniform guard: EXEC all-ones inside (WMMA requirement)
      const float* ap = aRow + kc * KTILE;
      const float* bp = &sW[kb * NC + col0 + lm];
#pragma unroll 2
      for (int kk = 0; kk < KTILE; kk += 4) {
        v2f a = *reinterpret_cast<const v2f*>(ap + kk);
#pragma unroll
        for (int s = 0; s < 4; ++s) {
          v2f b;
          b.x = bp[kk * NC + s * 16];
          b.y = bp[kk * NC + NC + s * 16];
          acc[s] = __builtin_amdgcn_wmma_f32_16x16x4_f32(
              false, a, false, b, (short)0, acc[s], false, false);
        }
      }
    }
    __syncthreads();
  }

  if (active) {
    float* cRow = C + (size_t)(row0 + rO) * LDC + col0 + lm;
#pragma unroll
    for (int v = 0; v < 8; ++v)
#pragma unroll
      for (int s = 0; s < 4; ++s)
        cRow[(size_t)v * LDC + s * 16] = acc[s][v];
  }
}

// split x(N,256) into stacked streams: cur[0..N) = x[:,128:256], cur[N..2N) = x[:,0:128]
__global__ void init_split(const float* __restrict__ x, float* __restrict__ cur) {
  int tid = blockIdx.x * blockDim.x + threadIdx.x;   // N*32 threads, float4 each
  if (tid >= NNODES * 32) return;
  int node = tid >> 5, q = (tid & 31) << 2;
  const float4 a = *reinterpret_cast<const float4*>(&x[(size_t)node * 256 + 128 + q]);
  const float4 b = *reinterpret_cast<const float4*>(&x[(size_t)node * 256 + q]);
  *reinterpret_cast<float4*>(&cur[(size_t)node * FDIM + q]) = a;
  *reinterpret_cast<float4*>(&cur[(size_t)(NNODES + node) * FDIM + q]) = b;
}

__global__ void zero_f4(float* __restrict__ p, int n4) {
  int i = blockIdx.x * blockDim.x + threadIdx.x;
  if (i < n4) reinterpret_cast<float4*>(p)[i] = make_float4(0.f, 0.f, 0.f, 0.f);
}

// one wave per (edge, direction); lane handles 4 of 128 columns
__global__ void scatter_edges(const float* __restrict__ t,
                              const long long* __restrict__ ei,
                              const float* __restrict__ ew,
                              float* __restrict__ agg) {
  int wid  = (blockIdx.x * blockDim.x + threadIdx.x) >> 5;
  int lane = threadIdx.x & 31;
  if (wid >= 2 * NEDGES) return;
  int strm = (wid >= NEDGES);
  int e    = strm ? wid - NEDGES : wid;
  int i0 = (int)ei[e];
  int i1 = (int)ei[NEDGES + e];
  // stream 0 (x1): gather from edge_index[1], scatter to edge_index[0]
  // stream 1 (x2): gather from edge_index[0], scatter to edge_index[1]
  int s = (strm ? i0 : i1) + strm * NNODES;
  int d = (strm ? i1 : i0) + strm * NNODES;
  float w = ew[e];
  int c = lane << 2;
  const float4 v = *reinterpret_cast<const float4*>(&t[(size_t)s * FDIM + c]);
  float* ap = &agg[(size_t)d * FDIM + c];
  unsafeAtomicAdd(ap + 0, w * v.x);
  unsafeAtomicAdd(ap + 1, w * v.y);
  unsafeAtomicAdd(ap + 2, w * v.z);
  unsafeAtomicAdd(ap + 3, w * v.w);
}

// cur = leaky_relu(agg + (self_loop ? t : 0))
__global__ void finish_layer(const float* __restrict__ agg, const float* __restrict__ t,
                             float* __restrict__ cur, int n4, int self_loop) {
  int i = blockIdx.x * blockDim.x + threadIdx.x;
  if (i >= n4) return;
  float4 a = reinterpret_cast<const float4*>(agg)[i];
  if (self_loop) {
    float4 b = reinterpret_cast<const float4*>(t)[i];
    a.x += b.x; a.y += b.y; a.z += b.z; a.w += b.w;
  }
  a.x = lrelu(a.x); a.y = lrelu(a.y); a.z = lrelu(a.z); a.w = lrelu(a.w);
  reinterpret_cast<float4*>(cur)[i] = a;
}

// out[n] = leaky_relu(dot(y[n,:256], fc2_W) + fc2_b); one wave per node
__global__ void fc2_head(const float* __restrict__ y, const float* __restrict__ w,
                         const float* __restrict__ b, float* __restrict__ out) {
  int tid = blockIdx.x * blockDim.x + threadIdx.x;
  int node = tid >> 5, lane = tid & 31;
  if (node >= NNODES) return;
  const float4* yr = reinterpret_cast<const float4*>(y + (size_t)node * 256);
  const float4* wr = reinterpret_cast<const float4*>(w);
  float4 a0 = yr[lane],      w0 = wr[lane];
  float4 a1 = yr[lane + 32], w1 = wr[lane + 32];
  float s = a0.x * w0.x + a0.y * w0.y + a0.z * w0.z + a0.w * w0.w
          + a1.x * w1.x + a1.y * w1.y + a1.z * w1.z + a1.w * w1.w;
#pragma unroll
  for (int m = 16; m; m >>= 1) s += __shfl_xor(s, m, 32);
  if (lane == 0) out[node] = lrelu(s + b[0]);
}

extern "C" void kernel_launch(void* const* d_in, const int* in_sizes, int n_in,
                              void* d_out, int out_size, void* d_ws, size_t ws_size,
                              hipStream_t stream) {
  const float*     x     = (const float*)d_in[0];
  const long long* ei    = (const long long*)d_in[1];
  const float*     ew    = (const float*)d_in[2];
  const float*     convW = (const float*)d_in[3];
  const float*     convB = (const float*)d_in[4];
  const float*     w1    = (const float*)d_in[5];
  const float*     w2    = (const float*)d_in[6];
  const float*     fc1W  = (const float*)d_in[7];
  const float*     fc1b  = (const float*)d_in[8];
  const float*     fc2W  = (const float*)d_in[9];
  const float*     fc2b  = (const float*)d_in[10];

  float* out = (float*)d_out;        // N
  float* y   = out + NNODES;         // N x 256 (second tuple element, written in place)

  const size_t half = (size_t)2 * NNODES * FDIM;   // 2N*128 floats per buffer
  float* cur = (float*)d_ws;
  float* t   = cur + half;
  float* agg = t + half;
  float* z   = t;                    // reuse t after the GCN layers

  const int n4 = (int)(half / 4);    // float4 count for 2N*128

  // split input columns into the two stacked streams
  init_split<<<(NNODES * 32 + 255) / 256, 256, 0, stream>>>(x, cur);

  // 3 GCN layers on both streams at once (M = 2N)
  const int convTiles = (2 * NNODES / 16) * (FDIM / 64);          // 12500
  const int scatBlocks = (2 * NEDGES * 32 + 255) / 256;           // 200000
  for (int i = 0; i < 3; ++i) {
    gemm_wmma<128, 128, 128, 128, 128><<<(convTiles + 7) / 8, 256, 0, stream>>>(
        cur, convW + (size_t)i * FDIM * FDIM, convB + (size_t)i * FDIM,
        t, 2 * NNODES);
    zero_f4<<<(n4 + 255) / 256, 256, 0, stream>>>(agg, n4);
    scatter_edges<<<scatBlocks, 256, 0, stream>>>(t, ei, ew, agg);
    finish_layer<<<(n4 + 255) / 256, 256, 0, stream>>>(agg, t, cur, n4, i > 0);
  }

  // z[:, 0:128] = x1 @ w1 ; z[:, 128:256] = x2 @ w2   (z strided at 256)
  const int projTiles = (NNODES / 16) * (FDIM / 64);              // 6250
  gemm_wmma<128, 128, 128, 128, 256><<<(projTiles + 7) / 8, 256, 0, stream>>>(
      cur, w1, nullptr, z, NNODES);
  gemm_wmma<128, 128, 128, 128, 256><<<(projTiles + 7) / 8, 256, 0, stream>>>(
      cur + (size_t)NNODES * FDIM, w2, nullptr, z + 128, NNODES);

  // y = z @ fc1_W + fc1_b   (K = 256, NC = 256, staged in 64-row LDS chunks)
  const int fc1Tiles = (NNODES / 16) * (256 / 64);                // 12500
  gemm_wmma<256, 64, 256, 256, 256><<<(fc1Tiles + 7) / 8, 256, 0, stream>>>(
      z, fc1W, fc1b, y, NNODES);

  // out = leaky_relu(y @ fc2_W + fc2_b)
  fc2_head<<<(NNODES * 32 + 255) / 256, 256, 0, stream>>>(y, fc2W, fc2b, out);
}